// Polynormer_62723702391576
// MI455X (gfx1250) — compile-verified
//
#include <hip/hip_runtime.h>
#include <hip/hip_bf16.h>

// Polynormer local block for MI455X (gfx1250, wave32, WMMA).
// Dense node GEMMs use V_WMMA_F32_16X16X4_F32 (exact fp32, matches reference
// numerics). Edge softmax/scatter uses L2-resident atomics; all feature maps
// (4 x 25.6MB) fit in the 192MB L2.

#define NN   50000
#define EE   800000
#define DD   128
#define LL   3
#define OUTD 40
#define LRELU 0.2f
#define LN_EPS 1e-5f
// order-preserving encoding of -inf: ~bits(0xFF800000)
#define ENC_NEG_INF 0x007FFFFFu

typedef float v2f __attribute__((ext_vector_type(2)));
typedef float v8f __attribute__((ext_vector_type(8)));

// ---------------------------------------------------------------------------
// GEMM: out[r, c] = sum_k A[r, k] * W[k, c]  (K fixed = 128, n_cols <= 128)
//   + optional bias[c] + optional addend[r, c], optional relu.
// Block = 256 threads = 8 waves, covers 32 rows x n_cols.
// Wave w owns 16-col tile w and BOTH 16-row halves (two accumulators sharing
// one B fragment -> 2 independent WMMA chains per LDS B load).
// A tile and transposed W both staged in LDS (pad 132 -> conflict-free b64).
// ---------------------------------------------------------------------------
__global__ __launch_bounds__(256) void wmma_gemm_kernel(
    const float* __restrict__ A, const float* __restrict__ W,
    const float* __restrict__ bias, const float* __restrict__ addend,
    float* __restrict__ out, int n_rows, int n_cols, int do_relu)
{
  __shared__ float As[32 * 132];    // A[row][k]
  __shared__ float Ws[128 * 132];   // W transposed: Ws[col][k]
  const int row0 = blockIdx.x * 32;
  const int tid  = threadIdx.x;

  // A tile: 32x128, coalesced
  for (int i = tid; i < 32 * 128; i += 256) {
    const int r = i >> 7, c = i & 127;
    As[r * 132 + c] = (row0 + r < n_rows) ? A[(size_t)(row0 + r) * DD + c] : 0.0f;
  }
  // W: 128(K) x n_cols, stored transposed; zero-fill cols >= n_cols
  for (int i = tid; i < 128 * 128; i += 256) {
    const int k = i >> 7, c = i & 127;
    Ws[c * 132 + k] = (c < n_cols) ? W[(size_t)k * n_cols + c] : 0.0f;
  }
  __syncthreads();

  const int wave = tid >> 5;         // 0..7 -> 16-col tile
  const int lane = tid & 31;
  const int col0 = wave * 16;
  if (col0 >= n_cols) return;        // idle waves (final 40-col projection)

  const int half = lane >> 4;        // 0: lanes 0-15, 1: lanes 16-31
  const int l16  = lane & 15;
  const int bcol = col0 + l16;

  // A frag layout: v0 = K=k (lanes 0-15) / K=k+2 (lanes 16-31); v1 = +1.
  // B frag mirrors A with N striped across lanes.
  const float* pa0 = As + (size_t)l16 * 132        + half * 2;
  const float* pa1 = As + (size_t)(16 + l16) * 132 + half * 2;
  const float* pb  = Ws + (size_t)bcol * 132       + half * 2;

  v8f acc0 = {0.f, 0.f, 0.f, 0.f, 0.f, 0.f, 0.f, 0.f};
  v8f acc1 = {0.f, 0.f, 0.f, 0.f, 0.f, 0.f, 0.f, 0.f};
#pragma unroll
  for (int k = 0; k < 128; k += 4) {
    const v2f a0 = *(const v2f*)(pa0 + k);
    const v2f a1 = *(const v2f*)(pa1 + k);
    const v2f b  = *(const v2f*)(pb  + k);
    acc0 = __builtin_amdgcn_wmma_f32_16x16x4_f32(
        false, a0, false, b, (short)0, acc0, false, false);
    acc1 = __builtin_amdgcn_wmma_f32_16x16x4_f32(
        false, a1, false, b, (short)0, acc1, false, false);
  }

  const bool colok   = (bcol < n_cols);
  const bool has_add = (addend != nullptr);
  const float bv = (bias && colok) ? bias[bcol] : 0.0f;
  if (!colok) return;
#pragma unroll
  for (int j = 0; j < 8; ++j) {
    const int m = half * 8 + j;      // C layout: vgpr j -> M=j (lo) / M=8+j (hi)
    const int r0 = row0 + m;
    const int r1 = row0 + 16 + m;
    if (r0 < n_rows) {
      float v = acc0[j] + bv;
      if (has_add) v += addend[(size_t)r0 * n_cols + bcol];
      if (do_relu) v = fmaxf(v, 0.0f);
      out[(size_t)r0 * n_cols + bcol] = v;
    }
    if (r1 < n_rows) {
      float v = acc1[j] + bv;
      if (has_add) v += addend[(size_t)r1 * n_cols + bcol];
      if (do_relu) v = fmaxf(v, 0.0f);
      out[(size_t)r1 * n_cols + bcol] = v;
    }
  }
}

// ---------------------------------------------------------------------------
// Per-node attention logits: a_src[n] = <xh[n,:], att_src>, same for dst.
// One wave32 per node, 4 channels per lane, shuffle reduction.
// ---------------------------------------------------------------------------
__global__ __launch_bounds__(256) void alpha_kernel(
    const float* __restrict__ xh, const float* __restrict__ att_s,
    const float* __restrict__ att_d, float* __restrict__ a_src,
    float* __restrict__ a_dst)
{
  const int node = blockIdx.x * 8 + (threadIdx.x >> 5);
  if (node >= NN) return;
  const int lane = threadIdx.x & 31;
  float s = 0.0f, d = 0.0f;
#pragma unroll
  for (int j = 0; j < 4; ++j) {
    const int c = lane + 32 * j;
    const float v = xh[(size_t)node * DD + c];
    s += v * att_s[c];
    d += v * att_d[c];
  }
  for (int m = 16; m > 0; m >>= 1) {
    s += __shfl_xor(s, m, 32);
    d += __shfl_xor(d, m, 32);
  }
  if (lane == 0) { a_src[node] = s; a_dst[node] = d; }
}

// ---------------------------------------------------------------------------
// Per-layer init: zero GAT accumulator, reset per-node max/denominator.
// ---------------------------------------------------------------------------
__global__ __launch_bounds__(256) void init_kernel(
    float* __restrict__ g, unsigned* __restrict__ nmax,
    float* __restrict__ nden)
{
  const size_t i = (size_t)blockIdx.x * blockDim.x + threadIdx.x;
  if (i < (size_t)NN * DD) g[i] = 0.0f;
  if (i < NN) { nmax[i] = ENC_NEG_INF; nden[i] = 0.0f; }
}

__device__ __forceinline__ unsigned fenc(float f) {
  const unsigned u = __float_as_uint(f);
  return (u & 0x80000000u) ? ~u : (u | 0x80000000u);
}
__device__ __forceinline__ float fdec(unsigned u) {
  const unsigned b = (u & 0x80000000u) ? (u & 0x7fffffffu) : ~u;
  return __uint_as_float(b);
}
__device__ __forceinline__ float lrelu(float e) {
  return e > 0.0f ? e : LRELU * e;
}

__global__ __launch_bounds__(256) void edge_max_kernel(
    const long long* __restrict__ src, const long long* __restrict__ dst,
    const float* __restrict__ a_src, const float* __restrict__ a_dst,
    unsigned* __restrict__ nmax)
{
  const int i = blockIdx.x * blockDim.x + threadIdx.x;
  if (i >= EE) return;
  const int s = (int)src[i], d = (int)dst[i];
  atomicMax(&nmax[d], fenc(lrelu(a_src[s] + a_dst[d])));
}

__global__ __launch_bounds__(256) void edge_expsum_kernel(
    const long long* __restrict__ src, const long long* __restrict__ dst,
    const float* __restrict__ a_src, const float* __restrict__ a_dst,
    const unsigned* __restrict__ nmax, float* __restrict__ exv,
    float* __restrict__ nden)
{
  const int i = blockIdx.x * blockDim.x + threadIdx.x;
  if (i >= EE) return;
  const int s = (int)src[i], d = (int)dst[i];
  const float e  = lrelu(a_src[s] + a_dst[d]);
  const float ex = expf(e - fdec(nmax[d]));
  exv[i] = ex;
  atomicAdd(&nden[d], ex);
}

// One wave32 per edge: msg = xh[src,:] * alpha scattered into g[dst,:].
__global__ __launch_bounds__(256) void edge_scatter_kernel(
    const long long* __restrict__ src, const long long* __restrict__ dst,
    const float* __restrict__ xh, const float* __restrict__ exv,
    const float* __restrict__ nden, float* __restrict__ g)
{
  const int e = blockIdx.x * 8 + (threadIdx.x >> 5);
  if (e >= EE) return;
  const int lane = threadIdx.x & 31;
  const int s = (int)src[e], d = (int)dst[e];
  const float alpha = exv[e] / (nden[d] + 1e-16f);
  const float4 v = *(const float4*)(xh + (size_t)s * DD + lane * 4);
  float* gp = g + (size_t)d * DD + lane * 4;
  atomicAdd(gp + 0, v.x * alpha);
  atomicAdd(gp + 1, v.y * alpha);
  atomicAdd(gp + 2, v.z * alpha);
  atomicAdd(gp + 3, v.w * alpha);
}

// ---------------------------------------------------------------------------
// Fused epilogue: v = post-relu combine (in g); w = h*v;
// x_new = (1-sigmoid(beta)) * (LN(w)*gamma+beta_ln) + sigmoid(beta) * v.
// One wave32 per node, two-pass LN entirely in registers.
// ---------------------------------------------------------------------------
__global__ __launch_bounds__(256) void fuse_ln_kernel(
    const float* __restrict__ h, const float* __restrict__ v,
    const float* __restrict__ ln_g, const float* __restrict__ ln_b,
    const float* __restrict__ betas, float* __restrict__ x)
{
  const int node = blockIdx.x * 8 + (threadIdx.x >> 5);
  if (node >= NN) return;
  const int lane = threadIdx.x & 31;
  const int c0 = lane * 4;

  const float4 h4 = *(const float4*)(h + (size_t)node * DD + c0);
  const float4 v4 = *(const float4*)(v + (size_t)node * DD + c0);
  float w[4] = {h4.x * v4.x, h4.y * v4.y, h4.z * v4.z, h4.w * v4.w};

  float sum = w[0] + w[1] + w[2] + w[3];
  for (int m = 16; m > 0; m >>= 1) sum += __shfl_xor(sum, m, 32);
  const float mu = sum * (1.0f / DD);

  float sq = 0.0f;
#pragma unroll
  for (int j = 0; j < 4; ++j) { const float t = w[j] - mu; sq += t * t; }
  for (int m = 16; m > 0; m >>= 1) sq += __shfl_xor(sq, m, 32);
  const float rs = rsqrtf(sq * (1.0f / DD) + LN_EPS);

  const float vv[4] = {v4.x, v4.y, v4.z, v4.w};
  float o[4];
#pragma unroll
  for (int j = 0; j < 4; ++j) {
    const int c = c0 + j;
    const float y = (w[j] - mu) * rs * ln_g[c] + ln_b[c];
    const float beta = 1.0f / (1.0f + expf(-betas[c]));
    o[j] = (1.0f - beta) * y + beta * vv[j];
  }
  *(float4*)(x + (size_t)node * DD + c0) = make_float4(o[0], o[1], o[2], o[3]);
}

// ---------------------------------------------------------------------------
extern "C" void kernel_launch(void* const* d_in, const int* in_sizes, int n_in,
                              void* d_out, int out_size, void* d_ws, size_t ws_size,
                              hipStream_t stream) {
  const float*     X0      = (const float*)d_in[0];
  const long long* EI      = (const long long*)d_in[1];   // int64 per reference
  const float*     W_in    = (const float*)d_in[2];
  const float*     b_in    = (const float*)d_in[3];
  const float*     W_h     = (const float*)d_in[4];
  const float*     b_h     = (const float*)d_in[5];
  const float*     W_gat   = (const float*)d_in[6];
  const float*     att_src = (const float*)d_in[7];
  const float*     att_dst = (const float*)d_in[8];
  const float*     W_l     = (const float*)d_in[9];
  const float*     b_l     = (const float*)d_in[10];
  const float*     ln_g    = (const float*)d_in[11];
  const float*     ln_b    = (const float*)d_in[12];
  const float*     betas   = (const float*)d_in[13];
  const float*     W_pred  = (const float*)d_in[14];
  const float*     b_pred  = (const float*)d_in[15];
  float* out = (float*)d_out;

  const long long* src = EI;
  const long long* dst = EI + EE;

  const size_t ND = (size_t)NN * DD;
  float*    x    = (float*)d_ws;
  float*    h    = x + ND;
  float*    xh   = h + ND;
  float*    g    = xh + ND;
  float*    asrc = g + ND;
  float*    adst = asrc + NN;
  unsigned* nmax = (unsigned*)(adst + NN);
  float*    nden = (float*)(nmax + NN);
  float*    exv  = nden + NN;

  const dim3 B(256);
  const dim3 Ggemm((NN + 31) / 32);
  const dim3 Gnode((NN + 7) / 8);
  const dim3 Gedge((EE + 255) / 256);
  const dim3 Gscat((EE + 7) / 8);
  const dim3 Ginit((unsigned)((ND + 255) / 256));

  // x = X0 @ W_in + b_in
  wmma_gemm_kernel<<<Ggemm, B, 0, stream>>>(X0, W_in, b_in, nullptr, x, NN, DD, 0);

  for (int i = 0; i < LL; ++i) {
    const size_t wo = (size_t)i * DD * DD;
    const size_t vo = (size_t)i * DD;
    // h = relu(x @ W_h[i] + b_h[i])
    wmma_gemm_kernel<<<Ggemm, B, 0, stream>>>(x, W_h + wo, b_h + vo, nullptr, h, NN, DD, 1);
    // xh = x @ W_gat[i]
    wmma_gemm_kernel<<<Ggemm, B, 0, stream>>>(x, W_gat + wo, nullptr, nullptr, xh, NN, DD, 0);
    // attention logits, softmax over incoming edges, weighted scatter
    alpha_kernel<<<Gnode, B, 0, stream>>>(xh, att_src + vo, att_dst + vo, asrc, adst);
    init_kernel<<<Ginit, B, 0, stream>>>(g, nmax, nden);
    edge_max_kernel<<<Gedge, B, 0, stream>>>(src, dst, asrc, adst, nmax);
    edge_expsum_kernel<<<Gedge, B, 0, stream>>>(src, dst, asrc, adst, nmax, exv, nden);
    edge_scatter_kernel<<<Gscat, B, 0, stream>>>(src, dst, xh, exv, nden, g);
    // g = relu(gat + x @ W_l[i] + b_l[i])   (epilogue-aliased addend)
    wmma_gemm_kernel<<<Ggemm, B, 0, stream>>>(x, W_l + wo, b_l + vo, g, g, NN, DD, 1);
    // x = (1-beta)*LN(h*g) + beta*g
    fuse_ln_kernel<<<Gnode, B, 0, stream>>>(h, g, ln_g + vo, ln_b + vo, betas + vo, x);
    // out (+)= x @ W_pred  (+ b_pred on first layer) -- folds away x_local
    wmma_gemm_kernel<<<Ggemm, B, 0, stream>>>(
        x, W_pred, (i == 0) ? b_pred : nullptr, (i == 0) ? nullptr : out,
        out, NN, OUTD, 0);
  }
}